// QLayer_65481071397308
// MI455X (gfx1250) — compile-verified
//
#include <hip/hip_runtime.h>

// CDNA5 (gfx1250) wave32 WMMA types: f32 16x16x4 -> A,B = 2 VGPRs/lane, C/D = 8 VGPRs/lane
typedef float v2f __attribute__((ext_vector_type(2)));
typedef float v8f __attribute__((ext_vector_type(8)));

#define NQ       14
#define DIM      16384          // 2^14 amplitudes
#define NTHREADS 256            // 8 wave32 waves
#define NLAYERS  3

__global__ __launch_bounds__(NTHREADS)
void qsim_kernel(const float* __restrict__ state,
                 const float* __restrict__ params,
                 float* __restrict__ out)
{
    __shared__ float sRe[DIM];            // 64 KB
    __shared__ float sIm[DIM];            // 64 KB
    __shared__ float sGr[256];            // 16x16 gate matrix, real
    __shared__ float sGi[256];            // 16x16 gate matrix, imag
    __shared__ float sRed[NQ * NTHREADS]; // reduction scratch (14 KB)

    const int tid  = threadIdx.x;
    const int b    = blockIdx.x;
    const int lane = tid & 31;
    const int wave = tid >> 5;          // 0..7
    const int half = (lane >> 4) & 1;   // lane half selects K-pairs / M-halves
    const int l16  = lane & 15;

    // ---- load real state into LDS, imag = 0 ----
    const float* src = state + (size_t)b * DIM;
    for (int t = tid; t < DIM; t += NTHREADS) { sRe[t] = src[t]; sIm[t] = 0.0f; }
    __syncthreads();

    for (int l = 0; l < NLAYERS; ++l) {
        for (int g = 0; g < 4; ++g) {
            // ---------------------------------------------------------------
            // Build G = kron of 4 single-qubit RY*RX unitaries (identity on
            // wires 10,11 for the last group). One entry per thread.
            // ---------------------------------------------------------------
            {
                const int mp = tid >> 4;     // row  (output 4-bit index)
                const int m  = tid & 15;     // col  (input 4-bit index)
                float ar = 1.0f, ai = 0.0f;
                #pragma unroll
                for (int q = 0; q < 4; ++q) {
                    const int i = (mp >> (3 - q)) & 1;
                    const int j = (m  >> (3 - q)) & 1;
                    const int wire = (g < 3) ? (g * 4 + q) : ((q >= 2) ? (10 + q) : -1);
                    float er, ei;
                    if (wire < 0) {                       // identity factor
                        er = (i == j) ? 1.0f : 0.0f; ei = 0.0f;
                    } else {
                        const float tx = 0.5f * params[(l * NQ + wire) * 2 + 0];
                        const float ty = 0.5f * params[(l * NQ + wire) * 2 + 1];
                        const float cx = __cosf(tx), sx = __sinf(tx);
                        const float cy = __cosf(ty), sy = __sinf(ty);
                        // U = RY(ty)*RX(tx)
                        if (i == 0 && j == 0)      { er =  cy * cx; ei =  sy * sx; }
                        else if (i == 0)           { er = -sy * cx; ei = -cy * sx; }
                        else if (j == 0)           { er =  sy * cx; ei = -cy * sx; }
                        else                       { er =  cy * cx; ei = -sy * sx; }
                    }
                    const float nr = ar * er - ai * ei;
                    const float ni = ar * ei + ai * er;
                    ar = nr; ai = ni;
                }
                sGr[tid] = ar;
                sGi[tid] = ai;
            }
            __syncthreads();

            // group address decomposition: amplitude(m, c) with m-stride 2^sh
            const int sh   = (g < 3) ? (10 - 4 * g) : 0;
            const int mask = (1 << sh) - 1;

            // ---- A operand (G columns) per WMMA K-chunk, loaded once ----
            float agr[4][2], agi[4][2];
            #pragma unroll
            for (int k = 0; k < 4; ++k)
                #pragma unroll
                for (int r = 0; r < 2; ++r) {
                    const int col = 4 * k + 2 * half + r;   // A[M=l16][K=col]
                    agr[k][r] = sGr[l16 * 16 + col];
                    agi[k][r] = sGi[l16 * 16 + col];
                }

            // ---- complex GEMM S' = G*S over 64 column-tiles, 8 per wave ----
            for (int t = 0; t < 8; ++t) {
                const int c    = (wave * 8 + t) * 16 + l16;        // this lane's column
                const int base = ((c >> sh) << (sh + 4)) | (c & mask);

                float br[4][2], bi[4][2];
                #pragma unroll
                for (int k = 0; k < 4; ++k)
                    #pragma unroll
                    for (int r = 0; r < 2; ++r) {
                        const int m = 4 * k + 2 * half + r;        // B[K=m][N=c]
                        const int a = base | (m << sh);
                        br[k][r] = sRe[a];
                        bi[k][r] = sIm[a];
                    }

                v8f accR = {}; v8f accI = {};
                #pragma unroll
                for (int k = 0; k < 4; ++k) {                      // Re += Gr*Sr
                    v2f av; av.x = agr[k][0]; av.y = agr[k][1];
                    v2f bv; bv.x = br[k][0];  bv.y = br[k][1];
                    accR = __builtin_amdgcn_wmma_f32_16x16x4_f32(false, av, false, bv, (short)0, accR, false, false);
                }
                #pragma unroll
                for (int k = 0; k < 4; ++k) {                      // Re += (-Gi)*Si
                    v2f av; av.x = -agi[k][0]; av.y = -agi[k][1];
                    v2f bv; bv.x = bi[k][0];   bv.y = bi[k][1];
                    accR = __builtin_amdgcn_wmma_f32_16x16x4_f32(false, av, false, bv, (short)0, accR, false, false);
                }
                #pragma unroll
                for (int k = 0; k < 4; ++k) {                      // Im += Gr*Si
                    v2f av; av.x = agr[k][0]; av.y = agr[k][1];
                    v2f bv; bv.x = bi[k][0];  bv.y = bi[k][1];
                    accI = __builtin_amdgcn_wmma_f32_16x16x4_f32(false, av, false, bv, (short)0, accI, false, false);
                }
                #pragma unroll
                for (int k = 0; k < 4; ++k) {                      // Im += Gi*Sr
                    v2f av; av.x = agi[k][0]; av.y = agi[k][1];
                    v2f bv; bv.x = br[k][0];  bv.y = br[k][1];
                    accI = __builtin_amdgcn_wmma_f32_16x16x4_f32(false, av, false, bv, (short)0, accI, false, false);
                }

                // D write-back: VGPR r holds row (r + 8*half), col c
                #pragma unroll
                for (int r = 0; r < 8; ++r) {
                    const int m = r + 8 * half;
                    const int a = base | (m << sh);
                    sRe[a] = accR[r];
                    sIm[a] = accI[r];
                }
            }
            __syncthreads();
        }

        // ---------------------------------------------------------------
        // CNOT chain (w=0..12, in order) == Gray-code gather:
        //   psi'[j] = psi[j ^ (j>>1)]   (prefix-XOR, wire 0 = MSB)
        // In-place via register staging (64 amplitudes per thread).
        // ---------------------------------------------------------------
        {
            float tmp[64];
            #pragma unroll
            for (int t = 0; t < 64; ++t) {
                const int j = tid + t * NTHREADS;
                tmp[t] = sRe[j ^ (j >> 1)];
            }
            __syncthreads();
            #pragma unroll
            for (int t = 0; t < 64; ++t) sRe[tid + t * NTHREADS] = tmp[t];
            __syncthreads();
            #pragma unroll
            for (int t = 0; t < 64; ++t) {
                const int j = tid + t * NTHREADS;
                tmp[t] = sIm[j ^ (j >> 1)];
            }
            __syncthreads();
            #pragma unroll
            for (int t = 0; t < 64; ++t) sIm[tid + t * NTHREADS] = tmp[t];
            __syncthreads();
        }
    }

    // ---- Z expectations: out[b][w] = sum_j p_j * (1 - 2*bit_w(j)) ----
    float acc[NQ];
    #pragma unroll
    for (int w = 0; w < NQ; ++w) acc[w] = 0.0f;
    for (int t = tid; t < DIM; t += NTHREADS) {
        const float p = sRe[t] * sRe[t] + sIm[t] * sIm[t];
        #pragma unroll
        for (int w = 0; w < NQ; ++w)
            acc[w] += ((t >> (13 - w)) & 1) ? -p : p;
    }
    #pragma unroll
    for (int w = 0; w < NQ; ++w) sRed[w * NTHREADS + tid] = acc[w];
    __syncthreads();
    for (int s = NTHREADS / 2; s > 0; s >>= 1) {
        if (tid < s) {
            #pragma unroll
            for (int w = 0; w < NQ; ++w)
                sRed[w * NTHREADS + tid] += sRed[w * NTHREADS + tid + s];
        }
        __syncthreads();
    }
    if (tid < NQ) out[b * NQ + tid] = sRed[tid * NTHREADS];
}

extern "C" void kernel_launch(void* const* d_in, const int* in_sizes, int n_in,
                              void* d_out, int out_size, void* d_ws, size_t ws_size,
                              hipStream_t stream)
{
    const float* state  = (const float*)d_in[0];   // (B, 16384) f32
    const float* params = (const float*)d_in[1];   // (3, 14, 2) f32
    float* out = (float*)d_out;                    // (B, 14) f32
    const int B = in_sizes[0] / DIM;               // 1024
    qsim_kernel<<<B, NTHREADS, 0, stream>>>(state, params, out);
}